// CustomGCNLayer_85306640433594
// MI455X (gfx1250) — compile-verified
//
#include <hip/hip_runtime.h>

typedef __attribute__((ext_vector_type(2))) float v2f;
typedef __attribute__((ext_vector_type(8))) float v8f;

#define D_H 128
#define LDS_STRIDE 132   // 128 + 4 pad: bank-conflict-free fragment reads

// ---------------------------------------------------------------------------
// Kernel 1: agg = bias (broadcast), zero BN stats accumulators
// ---------------------------------------------------------------------------
__global__ void init_kernel(float* __restrict__ agg, float* __restrict__ stats,
                            const float* __restrict__ b, unsigned total) {
    unsigned i = blockIdx.x * blockDim.x + threadIdx.x;
    if (i < 256u) stats[i] = 0.0f;          // sum[128], sumsq[128]
    if (i < total) agg[i] = b[i & (D_H - 1)];
}

// ---------------------------------------------------------------------------
// Kernel 2: xw = x @ W via V_WMMA_F32_16X16X4_F32
// Block = 256 threads (8 waves). Block computes 128 rows x 128 cols.
// X staged row-major (A fragments = contiguous v2f along K).
// W staged TRANSPOSED (Wt[n][k]) so B fragments are also contiguous v2f
// along K -> single ds_load_b64 into an even-aligned pair, no re-pairing movs.
// ---------------------------------------------------------------------------
__global__ __launch_bounds__(256, 2)
void gemm_kernel(const float* __restrict__ x, const float* __restrict__ W,
                 float* __restrict__ xw) {
    __shared__ float Wt[128 * LDS_STRIDE];   // Wt[n*132 + k] = W[k][n]
    __shared__ float Xl[128 * LDS_STRIDE];   // Xl[r*132 + k] = x[rowBase+r][k]

    const int    tid     = threadIdx.x;
    const size_t rowBase = (size_t)blockIdx.x * 128;

    // Cooperative staged loads (float4 global reads, fully coalesced).
    for (int i = tid * 4; i < 128 * 128; i += 256 * 4) {
        int r = i >> 7, c = i & 127;           // r = K row of W / row of x tile
        float4 w4 = *(const float4*)(W + i);
        Wt[(c + 0) * LDS_STRIDE + r] = w4.x;   // transpose on store (one-time)
        Wt[(c + 1) * LDS_STRIDE + r] = w4.y;
        Wt[(c + 2) * LDS_STRIDE + r] = w4.z;
        Wt[(c + 3) * LDS_STRIDE + r] = w4.w;
        float4 x4 = *(const float4*)(x + rowBase * D_H + i);
        *(float4*)(Xl + r * LDS_STRIDE + c) = x4;   // 528B stride keeps 16B align
    }
    __syncthreads();

    const int lane  = tid & 31;
    const int wave  = tid >> 5;
    const int m     = lane & 15;        // M (A) / N (B,D) index within tile
    const int halfk = (lane >> 4) * 2;  // K offset contributed by lane half
    const int mBase = wave * 16;        // this wave's 16-row strip

    v8f acc[8];
#pragma unroll
    for (int t = 0; t < 8; ++t) acc[t] = (v8f){0, 0, 0, 0, 0, 0, 0, 0};

#pragma unroll
    for (int k = 0; k < 128; k += 4) {
        // A fragment: a[j] = X[mBase+m][k + halfk + j]   (contiguous v2f)
        v2f a = *(const v2f*)(Xl + (mBase + m) * LDS_STRIDE + k + halfk);
#pragma unroll
        for (int t = 0; t < 8; ++t) {
            // B fragment: b[j] = W[k + halfk + j][t*16 + m] = Wt[t*16+m][k+halfk+j]
            v2f bfrag = *(const v2f*)(Wt + (t * 16 + m) * LDS_STRIDE + k + halfk);
            acc[t] = __builtin_amdgcn_wmma_f32_16x16x4_f32(
                false, a, false, bfrag, (short)0, acc[t], false, false);
        }
    }

    // D layout: lane n = lane%16, row M = v + 8*(lane/16)
#pragma unroll
    for (int t = 0; t < 8; ++t) {
#pragma unroll
        for (int v = 0; v < 8; ++v) {
            int M = v + 8 * (lane >> 4);
            xw[(rowBase + mBase + M) * D_H + t * 16 + m] = acc[t][v];
        }
    }
}

// ---------------------------------------------------------------------------
// Kernel 3: agg[dst] += edge_attr * xw[src]; one wave per edge, float4 lanes
// ---------------------------------------------------------------------------
__global__ void scatter_kernel(const float* __restrict__ xw,
                               const int* __restrict__ ei,
                               const float* __restrict__ ea,
                               float* __restrict__ agg, int E) {
    unsigned gid = blockIdx.x * blockDim.x + threadIdx.x;
    int e = (int)(gid >> 5);      // all 32 lanes of a wave share one edge
    int q = (int)(gid & 31);      // 4 features per lane
    if (e >= E) return;
    int   src = ei[e];
    int   dst = ei[E + e];
    float w   = ea[e];
    const float4 v = *(const float4*)(xw + (size_t)src * D_H + q * 4);
    float* p = agg + (size_t)dst * D_H + q * 4;
    atomicAdd(p + 0, v.x * w);
    atomicAdd(p + 1, v.y * w);
    atomicAdd(p + 2, v.z * w);
    atomicAdd(p + 3, v.w * w);
}

// ---------------------------------------------------------------------------
// Kernel 4: per-feature sum / sumsq partial reduction
// ---------------------------------------------------------------------------
__global__ void stats_kernel(const float* __restrict__ agg,
                             float* __restrict__ stats, int N) {
    int f    = threadIdx.x & (D_H - 1);
    int half = threadIdx.x >> 7;  // two node-rows in flight per block
    float s = 0.0f, ss = 0.0f;
    for (int n = blockIdx.x * 2 + half; n < N; n += gridDim.x * 2) {
        float v = agg[(size_t)n * D_H + f];
        s += v;
        ss += v * v;
    }
    __shared__ float ls[256], lss[256];
    ls[threadIdx.x]  = s;
    lss[threadIdx.x] = ss;
    __syncthreads();
    if (half == 0) {
        s  += ls[128 + f];
        ss += lss[128 + f];
        atomicAdd(&stats[f], s);
        atomicAdd(&stats[D_H + f], ss);
    }
}

// ---------------------------------------------------------------------------
// Kernel 5: BN (training stats) + affine + ReLU, in place on d_out
// ---------------------------------------------------------------------------
__global__ void bn_relu_kernel(float* __restrict__ out,
                               const float* __restrict__ stats,
                               const float* __restrict__ gamma,
                               const float* __restrict__ beta,
                               unsigned total, float invN) {
    unsigned i = blockIdx.x * blockDim.x + threadIdx.x;
    if (i >= total) return;
    int   f    = i & (D_H - 1);
    float mean = stats[f] * invN;
    float var  = stats[D_H + f] * invN - mean * mean;
    float xn   = (out[i] - mean) * rsqrtf(var + 1e-5f);
    float r    = gamma[f] * xn + beta[f];
    out[i] = fmaxf(r, 0.0f);
}

// ---------------------------------------------------------------------------
extern "C" void kernel_launch(void* const* d_in, const int* in_sizes, int n_in,
                              void* d_out, int out_size, void* d_ws, size_t ws_size,
                              hipStream_t stream) {
    const float* x     = (const float*)d_in[0];
    const int*   ei    = (const int*)d_in[1];
    const float* ea    = (const float*)d_in[2];
    const int*   batch = (const int*)d_in[3];
    const float* W     = (const float*)d_in[4];
    const float* b     = (const float*)d_in[5];
    const float* gamma = (const float*)d_in[6];
    const float* beta  = (const float*)d_in[7];

    const int      N     = in_sizes[0] / D_H;   // 64000
    const int      E     = in_sizes[2];         // 640000
    const unsigned total = (unsigned)N * D_H;   // 8,192,000

    float* xw    = (float*)d_ws;                // [N, 128] = 32.8 MB scratch
    float* stats = xw + total;                  // 256 floats
    float* out   = (float*)d_out;               // agg -> normalized output

    init_kernel<<<(total + 255) / 256, 256, 0, stream>>>(out, stats, b, total);
    gemm_kernel<<<N / 128, 256, 0, stream>>>(x, W, xw);
    scatter_kernel<<<((unsigned)E * 32 + 255) / 256, 256, 0, stream>>>(xw, ei, ea, out, E);
    stats_kernel<<<256, 256, 0, stream>>>(out, stats, N);
    bn_relu_kernel<<<(total + 255) / 256, 256, 0, stream>>>(out, stats, gamma, beta,
                                                            total, 1.0f / (float)N);

    // Tail of the output tuple: edge_index (int32), edge_attr (f32), batch (int32)
    float* tail = out + total;
    hipMemcpyAsync(tail, ei, sizeof(int) * 2 * (size_t)E, hipMemcpyDeviceToDevice, stream);
    hipMemcpyAsync(tail + 2 * (size_t)E, ea, sizeof(float) * (size_t)E,
                   hipMemcpyDeviceToDevice, stream);
    hipMemcpyAsync(tail + 3 * (size_t)E, batch, sizeof(int) * (size_t)N,
                   hipMemcpyDeviceToDevice, stream);
}